// ComplexGCN_29403346108560
// MI455X (gfx1250) — compile-verified
//
#include <hip/hip_runtime.h>

typedef float v2f __attribute__((ext_vector_type(2)));
typedef float v8f __attribute__((ext_vector_type(8)));

#define TPB 256
#define BN_EPS 1e-5f

// ---------------- degree / normalization ----------------
__global__ void k_deg_init(float* __restrict__ deg, int N) {
  int i = blockIdx.x * blockDim.x + threadIdx.x;
  if (i < N) deg[i] = 1.0f;  // +1 for self loop
}

__global__ void k_deg_count(const int* __restrict__ dst, float* __restrict__ deg, int E) {
  int e = blockIdx.x * blockDim.x + threadIdx.x;
  if (e < E) atomicAdd(&deg[dst[e]], 1.0f);
}

__global__ void k_dinv(const float* __restrict__ deg, float* __restrict__ dinv, int N) {
  int i = blockIdx.x * blockDim.x + threadIdx.x;
  if (i < N) dinv[i] = rsqrtf(deg[i]);
}

// ---------------- dense GEMM via fp32 WMMA ----------------
// Out[N, NT*16] = A[N, K] x W[K, NT*16]. One wave computes a 16-row strip
// across all NT column tiles, reusing the A fragment each K-step.
template <int NT>
__global__ void k_gemm_wmma(const float* __restrict__ A,
                            const float* __restrict__ W,
                            float* __restrict__ Out,
                            int N, int K) {
  const int Fo   = NT * 16;
  const int gtid = blockIdx.x * blockDim.x + threadIdx.x;
  const int wave = gtid >> 5;
  const int lane = threadIdx.x & 31;
  const int m0   = wave * 16;
  if (m0 >= N) return;

  const int half = lane >> 4;   // 0: K pair {0,1}, 1: K pair {2,3}
  const int l16  = lane & 15;   // M index for A, N index for B/C/D

  v8f acc[NT];
#pragma unroll
  for (int t = 0; t < NT; ++t) acc[t] = (v8f){0.f, 0.f, 0.f, 0.f, 0.f, 0.f, 0.f, 0.f};

  const float* arow = A + (size_t)(m0 + l16) * K;

  for (int k0 = 0; k0 < K; k0 += 4) {
    const int ka = k0 + half * 2;
    // A fragment: lane holds A[m0+l16, ka] and A[m0+l16, ka+1]
    v2f a = *(const v2f*)(arow + ka);
#pragma unroll
    for (int t = 0; t < NT; ++t) {
      const int n = t * 16 + l16;
      v2f b;
      b.x = W[(size_t)ka * Fo + n];
      b.y = W[(size_t)(ka + 1) * Fo + n];
      acc[t] = __builtin_amdgcn_wmma_f32_16x16x4_f32(
          /*neg_a=*/false, a, /*neg_b=*/false, b,
          /*c_mod=*/(short)0, acc[t], /*reuse_a=*/false, /*reuse_b=*/false);
    }
  }

  // C/D layout: VGPR r -> M = r + 8*half, N = l16
#pragma unroll
  for (int t = 0; t < NT; ++t) {
#pragma unroll
    for (int r = 0; r < 8; ++r) {
      const int m = m0 + r + half * 8;
      Out[(size_t)m * Fo + t * 16 + l16] = acc[t][r];
    }
  }
}

// ---------------- self-loop init: agg = xw / deg ----------------
__global__ void k_selfloop(const float* __restrict__ xw, const float* __restrict__ deg,
                           float* __restrict__ agg, int N, int F) {
  int i = blockIdx.x * blockDim.x + threadIdx.x;
  if (i < N * F) {
    int node = i / F;
    agg[i] = xw[i] / deg[node];
  }
}

// ---------------- edge scatter: agg[dst] += xw[src] * dinv[s]*dinv[d] ----------------
// One thread per (edge, 4-float chunk); consecutive threads cover consecutive
// features of the same src row -> coalesced float4 gathers; xw (<=51MB) is
// L2-resident so gathers/atomics run at L2 bandwidth.
__global__ void k_edge_agg(const float* __restrict__ xw,
                           const int* __restrict__ src, const int* __restrict__ dst,
                           const float* __restrict__ dinv,
                           float* __restrict__ agg, int E, int F) {
  const int chunks = F >> 2;
  long long t = (long long)blockIdx.x * blockDim.x + threadIdx.x;
  long long total = (long long)E * chunks;
  if (t >= total) return;
  int e = (int)(t / chunks);
  int c = (int)(t % chunks);
  int s = src[e], d = dst[e];
  float norm = dinv[s] * dinv[d];
  float4 v = ((const float4*)(xw + (size_t)s * F))[c];
  float* out = agg + (size_t)d * F + (size_t)c * 4;
  atomicAdd(out + 0, v.x * norm);
  atomicAdd(out + 1, v.y * norm);
  atomicAdd(out + 2, v.z * norm);
  atomicAdd(out + 3, v.w * norm);
}

// ---------------- fused bias + BN + ReLU (in place) ----------------
__global__ void k_bias_bn_relu(float* __restrict__ h, const float* __restrict__ bias,
                               const float* __restrict__ g, const float* __restrict__ be,
                               const float* __restrict__ m, const float* __restrict__ v,
                               int N, int F) {
  int i = blockIdx.x * blockDim.x + threadIdx.x;
  if (i < N * F) {
    int f = i % F;
    float x = h[i] + bias[f];
    x = (x - m[f]) * rsqrtf(v[f] + BN_EPS) * g[f] + be[f];
    h[i] = fmaxf(x, 0.0f);
  }
}

__global__ void k_bias(float* __restrict__ h, const float* __restrict__ bias, int N, int F) {
  int i = blockIdx.x * blockDim.x + threadIdx.x;
  if (i < N * F) h[i] += bias[i % F];
}

// ---------------- orchestration ----------------
static inline int cdiv(long long a, int b) { return (int)((a + b - 1) / b); }

extern "C" void kernel_launch(void* const* d_in, const int* in_sizes, int n_in,
                              void* d_out, int out_size, void* d_ws, size_t ws_size,
                              hipStream_t stream) {
  const int F_IN = 128;
  const int N = in_sizes[0] / F_IN;
  const int E = in_sizes[1] / 2;

  const float* x = (const float*)d_in[0];
  const int* edge = (const int*)d_in[1];
  const int* src = edge;       // edge_index[0, :]
  const int* dst = edge + E;   // edge_index[1, :]

  const float* Wm[5], *Bv[5];
  for (int l = 0; l < 5; ++l) {
    Wm[l] = (const float*)d_in[2 + 2 * l];
    Bv[l] = (const float*)d_in[3 + 2 * l];
  }
  const float* Gv[4], *BEv[4], *Mv[4], *Vv[4];
  for (int l = 0; l < 4; ++l) {
    Gv[l]  = (const float*)d_in[12 + 4 * l];
    BEv[l] = (const float*)d_in[13 + 4 * l];
    Mv[l]  = (const float*)d_in[14 + 4 * l];
    Vv[l]  = (const float*)d_in[15 + 4 * l];
  }

  const int dims[6] = {128, 64, 128, 64, 32, 16};

  // workspace: deg[N] | dinv[N] | bufA[N*128] (xw) | bufB[N*128] (agg/h)
  float* deg  = (float*)d_ws;
  float* dinv = deg + N;
  float* bufA = dinv + N;
  float* bufB = bufA + (size_t)N * 128;

  // ---- degrees ----
  k_deg_init<<<cdiv(N, TPB), TPB, 0, stream>>>(deg, N);
  k_deg_count<<<cdiv(E, TPB), TPB, 0, stream>>>(dst, deg, E);
  k_dinv<<<cdiv(N, TPB), TPB, 0, stream>>>(deg, dinv, N);

  const float* cur = x;  // layer input
  for (int l = 0; l < 5; ++l) {
    const int K = dims[l];
    const int F = dims[l + 1];
    float* xw  = bufA;
    float* agg = (l == 4) ? (float*)d_out : bufB;

    // GEMM: xw = cur @ W
    {
      const int waves  = (N + 15) / 16;
      const int blocks = cdiv((long long)waves * 32, TPB);
      switch (F / 16) {
        case 1: k_gemm_wmma<1><<<blocks, TPB, 0, stream>>>(cur, Wm[l], xw, N, K); break;
        case 2: k_gemm_wmma<2><<<blocks, TPB, 0, stream>>>(cur, Wm[l], xw, N, K); break;
        case 4: k_gemm_wmma<4><<<blocks, TPB, 0, stream>>>(cur, Wm[l], xw, N, K); break;
        default: k_gemm_wmma<8><<<blocks, TPB, 0, stream>>>(cur, Wm[l], xw, N, K); break;
      }
    }

    // agg = xw/deg  (self loop), then scatter-add over edges
    k_selfloop<<<cdiv((long long)N * F, TPB), TPB, 0, stream>>>(xw, deg, agg, N, F);
    k_edge_agg<<<cdiv((long long)E * (F >> 2), TPB), TPB, 0, stream>>>(
        xw, src, dst, dinv, agg, E, F);

    if (l < 4) {
      k_bias_bn_relu<<<cdiv((long long)N * F, TPB), TPB, 0, stream>>>(
          agg, Bv[l], Gv[l], BEv[l], Mv[l], Vv[l], N, F);
      cur = agg;  // bufB becomes next layer's input
    } else {
      k_bias<<<cdiv((long long)N * F, TPB), TPB, 0, stream>>>(agg, Bv[l], N, F);
    }
  }
}